// NGCF_Embedding_5566277616503
// MI455X (gfx1250) — compile-verified
//
#include <hip/hip_runtime.h>

typedef __attribute__((ext_vector_type(2))) float v2f;
typedef __attribute__((ext_vector_type(8))) float v8f;

#define N_NODES 150000
#define N_EDGES 2400000
#define DIM     128
#define SLOPE   0.2f
#define L2EPS   1e-12f

// ---------------------------------------------------------------------------
// Kernel 1: zero the SpMM accumulator (d_ws is poisoned, must clear each call)
// ---------------------------------------------------------------------------
__global__ void zero_kernel(float4* __restrict__ p, int n4) {
    int i = blockIdx.x * blockDim.x + threadIdx.x;
    if (i < n4) p[i] = make_float4(0.f, 0.f, 0.f, 0.f);
}

// ---------------------------------------------------------------------------
// Kernel 2: SpMM  side[row] += val * ego[col]   (one wave32 per edge)
// lane l handles columns 4l..4l+3 -> coalesced 512B gather per wave,
// 4x global_atomic_add_f32 scatter (L2-resident: ego fits in 192MB L2).
// ---------------------------------------------------------------------------
__global__ __launch_bounds__(256)
void spmm_kernel(const int* __restrict__ erow, const int* __restrict__ ecol,
                 const float* __restrict__ eval, const float* __restrict__ ego,
                 float* __restrict__ side) {
    int edge = blockIdx.x * 8 + (threadIdx.x >> 5);   // 8 waves per block
    int lane = threadIdx.x & 31;
    if (edge >= N_EDGES) return;
    int r = erow[edge];          // wave-uniform -> scalar loads
    int c = ecol[edge];
    float v = eval[edge];
    float4 g = ((const float4*)(ego + (size_t)c * DIM))[lane];
    float* dst = side + (size_t)r * DIM + lane * 4;
    atomicAdd(dst + 0, v * g.x);
    atomicAdd(dst + 1, v * g.y);
    atomicAdd(dst + 2, v * g.z);
    atomicAdd(dst + 3, v * g.w);
}

// ---------------------------------------------------------------------------
// Kernel 3: fused dual-GEMM + bias + leaky_relu + add + L2 normalize.
// One block = 16-row tile. 8 waves, wave w owns output columns [16w,16w+16).
// Each wave runs two chained V_WMMA_F32_16X16X4_F32 accumulations over K=128.
//
// Fragment layouts (ISA 7.12.2, wave32):
//  A 16x4 f32 : v0 = A[M=lane%16][K = 2*(lane/16)+0], v1 = K+1
//  B  4x16 f32: v0 = B[K = 2*(lane/16)+0][N = lane%16], v1 = K+1
//  C/D 16x16  : vgpr i -> M = i + 8*(lane/16), N = lane%16
// ---------------------------------------------------------------------------
__global__ __launch_bounds__(256)
void fused_kernel(const float* __restrict__ side, const float* __restrict__ ego,
                  const float* __restrict__ wgc, const float* __restrict__ bgc,
                  const float* __restrict__ wbi, const float* __restrict__ bbi,
                  float* __restrict__ out) {
    __shared__ float s_side[16][DIM];
    __shared__ float s_ego[16][DIM];
    __shared__ float rowsq[16];

    const int tid = threadIdx.x;
    const int r0  = blockIdx.x * 16;           // N_NODES % 16 == 0

    // Stage both 16x128 tiles into LDS (float4, fully coalesced).
    {
        const float4* s4 = (const float4*)(side + (size_t)r0 * DIM);
        const float4* e4 = (const float4*)(ego  + (size_t)r0 * DIM);
        float4* ls = (float4*)&s_side[0][0];
        float4* le = (float4*)&s_ego[0][0];
        for (int i = tid; i < 16 * DIM / 4; i += 256) { ls[i] = s4[i]; le[i] = e4[i]; }
    }
    if (tid < 16) rowsq[tid] = 0.f;
    __syncthreads();

    const int wv   = tid >> 5;      // column tile 0..7
    const int lane = tid & 31;
    const int half = lane >> 4;     // 0: lanes 0-15, 1: lanes 16-31
    const int lm   = lane & 15;
    const int n    = wv * 16 + lm;  // output column of this lane

    v8f cg = {};   // side @ w_gc accumulator
    v8f cb = {};   // (ego*side) @ w_bi accumulator

    #pragma unroll 4
    for (int kt = 0; kt < DIM / 4; ++kt) {
        const int kb = kt * 4 + half * 2;
        float s0 = s_side[lm][kb], s1 = s_side[lm][kb + 1];
        float e0 = s_ego[lm][kb],  e1 = s_ego[lm][kb + 1];

        v2f ag; ag.x = s0;       ag.y = s1;        // A frag for GEMM 1
        v2f ab; ab.x = s0 * e0;  ab.y = s1 * e1;   // A frag for GEMM 2 (bi)
        v2f bg; bg.x = wgc[(size_t)kb * DIM + n];  bg.y = wgc[(size_t)(kb + 1) * DIM + n];
        v2f bb; bb.x = wbi[(size_t)kb * DIM + n];  bb.y = wbi[(size_t)(kb + 1) * DIM + n];

        cg = __builtin_amdgcn_wmma_f32_16x16x4_f32(false, ag, false, bg,
                                                   (short)0, cg, false, false);
        cb = __builtin_amdgcn_wmma_f32_16x16x4_f32(false, ab, false, bb,
                                                   (short)0, cb, false, false);
    }

    // Epilogue: bias + leaky_relu + add; butterfly-reduce row sum-of-squares.
    const float biasg = bgc[n];
    const float biasb = bbi[n];
    float ev[8];
    #pragma unroll
    for (int i = 0; i < 8; ++i) {
        float a = cg[i] + biasg;  a = a > 0.f ? a : SLOPE * a;
        float b = cb[i] + biasb;  b = b > 0.f ? b : SLOPE * b;
        float e = a + b;
        ev[i] = e;
        float s = e * e;
        // reduce over the 16 lanes of this half (xor<16 never crosses bit 4)
        for (int off = 1; off < 16; off <<= 1) s += __shfl_xor(s, off, 32);
        if (lm == 0) atomicAdd(&rowsq[half * 8 + i], s);
    }
    __syncthreads();

    #pragma unroll
    for (int i = 0; i < 8; ++i) {
        const int m = half * 8 + i;
        const float scale = rsqrtf(fmaxf(rowsq[m], L2EPS));
        out[(size_t)(r0 + m) * DIM + n] = ev[i] * scale;
    }
}

// ---------------------------------------------------------------------------
extern "C" void kernel_launch(void* const* d_in, const int* in_sizes, int n_in,
                              void* d_out, int out_size, void* d_ws, size_t ws_size,
                              hipStream_t stream) {
    const int*   erow = (const int*)  d_in[0];
    const int*   ecol = (const int*)  d_in[1];
    const float* eval = (const float*)d_in[2];
    const float* ego  = (const float*)d_in[3];
    const float* wgc  = (const float*)d_in[4];
    const float* bgc  = (const float*)d_in[5];
    const float* wbi  = (const float*)d_in[6];
    const float* bbi  = (const float*)d_in[7];
    float* out  = (float*)d_out;
    float* side = (float*)d_ws;                 // N_NODES*DIM floats (76.8 MB)

    const int n4 = N_NODES * DIM / 4;
    zero_kernel<<<(n4 + 255) / 256, 256, 0, stream>>>((float4*)side, n4);

    const int spmm_blocks = (N_EDGES + 7) / 8;  // 8 waves (edges) per block
    spmm_kernel<<<spmm_blocks, 256, 0, stream>>>(erow, ecol, eval, ego, side);

    fused_kernel<<<N_NODES / 16, 256, 0, stream>>>(side, ego, wgc, bgc, wbi, bbi, out);
}